// DotPooling_53678501265805
// MI455X (gfx1250) — compile-verified
//
#include <hip/hip_runtime.h>

typedef __attribute__((ext_vector_type(2))) float v2f;
typedef __attribute__((ext_vector_type(4))) float v4f;
typedef __attribute__((ext_vector_type(8))) float v8f;

#define N_ROWS   64
#define N_COLS   256
#define LDS_PITCH 260   // 256 + 4-float pad: conflict-free ds_load_b64, keeps 16B align
#define THREADS  320    // 10 waves (wave32): one wave per lower-tri 16x16 tile
#define OUT_PER_B (N_ROWS * (N_ROWS - 1) / 2)   // 2016

__global__ __launch_bounds__(THREADS)
void gram_tril_wmma_f32(const float* __restrict__ x, float* __restrict__ out) {
    extern __shared__ float xs[];   // 64 * 260 * 4 = 66560 B
    const int b   = blockIdx.x;
    const int tid = threadIdx.x;
    const float* __restrict__ xb = x + (size_t)b * (N_ROWS * N_COLS);

    // ---- Stage x[b] (64 KB) into LDS: exactly one HBM read per batch ----
    for (int idx = tid; idx < N_ROWS * (N_COLS / 4); idx += THREADS) {
        const int row = idx >> 6;        // 64 float4s per row
        const int c4  = idx & 63;
        v4f v = *(const v4f*)(xb + row * N_COLS + c4 * 4);
        *(v4f*)(&xs[row * LDS_PITCH + c4 * 4]) = v;
    }
    __syncthreads();

    // ---- Tile assignment: wave -> (ti, tj), ti >= tj, in the 4x4 tile grid ----
    const int wave = tid >> 5;
    const int lane = tid & 31;
    static const int TI[10] = {0,1,1,2,2,2,3,3,3,3};
    static const int TJ[10] = {0,0,1,0,1,2,0,1,2,3};
    const int ti = TI[wave];
    const int tj = TJ[wave];

    // A(16x4 f32) and B(4x16 f32) operands use the same per-lane gather from X:
    // lane L: rows tile*16+(L&15), cols k + ((L>=16)?2:0) .. +1  (8B each)
    const int lrow = lane & 15;
    const int koff = (lane >> 4) << 1;           // 0 or 2
    const float* aBase = &xs[(ti * 16 + lrow) * LDS_PITCH + koff];
    const float* bBase = &xs[(tj * 16 + lrow) * LDS_PITCH + koff];

    v8f acc = {0.f, 0.f, 0.f, 0.f, 0.f, 0.f, 0.f, 0.f};
    #pragma unroll 8
    for (int k = 0; k < N_COLS; k += 4) {
        v2f av = *(const v2f*)(aBase + k);
        v2f bv = *(const v2f*)(bBase + k);
        // (neg_a, A, neg_b, B, c_mod, C, reuse_a, reuse_b)
        acc = __builtin_amdgcn_wmma_f32_16x16x4_f32(
            false, av, false, bv, (short)0, acc, false, false);
    }

    // ---- Epilogue: out[b][g*(g-1)/2 + gj] for g > gj ----
    // D layout: VGPR v -> M = v + (lane>=16 ? 8 : 0), N = lane & 15.
    // Each element's index computed independently in unsigned 32-bit so the
    // backend can use scale_offset global stores (no 64-bit address chains).
    float* __restrict__ outb = out + (size_t)b * OUT_PER_B;
    const unsigned mhalf = (unsigned)((lane >> 4) << 3);   // 0 or 8
    const unsigned gbase = (unsigned)(ti * 16) + mhalf;    // row of acc[0]
    const unsigned gj    = (unsigned)(tj * 16 + lrow);

    if (ti != tj) {
        // Off-diagonal tile: g > gj always -> branch-free stores.
        #pragma unroll
        for (int v = 0; v < 8; ++v) {
            const unsigned g = gbase + (unsigned)v;
            outb[(g * (g - 1u)) / 2u + gj] = acc[v];
        }
    } else {
        // Diagonal tile: strictly-lower predicate per element.
        #pragma unroll
        for (int v = 0; v < 8; ++v) {
            const unsigned g = gbase + (unsigned)v;
            if (g > gj) outb[(g * (g - 1u)) / 2u + gj] = acc[v];
        }
    }
}

extern "C" void kernel_launch(void* const* d_in, const int* in_sizes, int n_in,
                              void* d_out, int out_size, void* d_ws, size_t ws_size,
                              hipStream_t stream) {
    const float* x = (const float*)d_in[0];
    float* out = (float*)d_out;
    const int B = in_sizes[0] / (N_ROWS * N_COLS);   // 4096
    const size_t lds_bytes = (size_t)N_ROWS * LDS_PITCH * sizeof(float);
    gram_tril_wmma_f32<<<B, THREADS, lds_bytes, stream>>>(x, out);
}